// Arima_0_1_1_25623774888069
// MI455X (gfx1250) — compile-verified
//
#include <hip/hip_runtime.h>

// ARIMA(0,1,1) innovations: tridiagonal Toeplitz solve via parallel cyclic
// reduction, fully LDS-resident in a single WGP (1024 threads = 32 wave32s).
//
// Problem size: N = 4096 interior unknowns, output 4097.
// Total global traffic ~32 KB -> entirely latency bound on MI455X
// (23.3 TB/s HBM => ~1.4 ns of bandwidth time). The win is one async
// global->LDS staging pass, 12 barrier-separated PCR sweeps in LDS,
// one store pass out.

#define AS1 __attribute__((address_space(1)))
#define AS3 __attribute__((address_space(3)))

typedef int v4i __attribute__((ext_vector_type(4)));

#define NPCR 4096   // number of tridiagonal unknowns
#define TPB  1024   // 32 waves on one WGP
#define PER_T (NPCR / TPB)

#if defined(__AMDGCN__) && \
    __has_builtin(__builtin_amdgcn_global_load_async_to_lds_b128) && \
    __has_builtin(__builtin_amdgcn_global_load_async_to_lds_b32)
#define HAVE_ASYNC_LDS 1
#else
#define HAVE_ASYNC_LDS 0
#endif

__device__ __forceinline__ void wait_async_zero() {
#if defined(__AMDGCN__)
#if __has_builtin(__builtin_amdgcn_s_wait_asynccnt)
    __builtin_amdgcn_s_wait_asynccnt(0);
#else
    asm volatile("s_wait_asynccnt 0" ::: "memory");
#endif
#endif
}

__global__ __launch_bounds__(TPB) void arima_ma1_innov_kernel(
    const float* __restrict__ time_block,   // 4097
    const float* __restrict__ arma_const,   // 1
    const float* __restrict__ ma_coeff,     // 1
    const float* __restrict__ std_innov,    // 1
    float* __restrict__ out)                // 4097
{
    // Layout: sa | sd | se | sr  (sub, diag, super, rhs) = 64 KB exactly.
    __shared__ float smem[4 * NPCR];
    float* sa = smem;
    float* sd = smem + NPCR;
    float* se = smem + 2 * NPCR;
    float* sr = smem + 3 * NPCR;

    const int t = threadIdx.x;

    // ---- Stage time_block[0..4095] into the sa region and time_block[4096]
    // ---- into sd[0] (i.e. smem[4096]) so smem[0..4096] holds the series.
#if HAVE_ASYNC_LDS
    {
        const AS1 float* g = (const AS1 float*)time_block;
        AS3 float* l = (AS3 float*)smem;
        // 1024 lanes x 16B = 4096 floats, one async b128 each.
        __builtin_amdgcn_global_load_async_to_lds_b128(
            (AS1 v4i*)(g + t * 4), (AS3 v4i*)(l + t * 4), 0, 0);
        if (t == 0) {
            __builtin_amdgcn_global_load_async_to_lds_b32(
                (AS1 int*)(g + NPCR), (AS3 int*)(l + NPCR), 0, 0);
        }
        wait_async_zero();
    }
#else
    for (int i = t; i < NPCR; i += TPB) smem[i] = time_block[i];
    if (t == 0) smem[NPCR] = time_block[NPCR];
#endif
    __syncthreads();

    const float c    = ma_coeff[0];
    const float cst  = arma_const[0];
    const float istd = 1.0f / std_innov[0];
    const float dia  = c * c + 1.0f;

    // rhs_i = time[i+1] - time[i] - const  (read staged data into registers)
    float rr[PER_T];
#pragma unroll
    for (int k = 0; k < PER_T; ++k) {
        int i = t + k * TPB;
        rr[k] = smem[i + 1] - smem[i] - cst;
    }
    __syncthreads();   // everyone done reading staged series; safe to overwrite

    // Initialize tridiagonal system: T = (c^2+1) I + c (S + S^T)
#pragma unroll
    for (int k = 0; k < PER_T; ++k) {
        int i = t + k * TPB;
        sa[i] = (i == 0)        ? 0.0f : c;
        se[i] = (i == NPCR - 1) ? 0.0f : c;
        sd[i] = dia;
        sr[i] = rr[k];
    }
    __syncthreads();

    // ---- Parallel cyclic reduction: 12 levels, strides 1..2048.
    for (int s = 1; s < NPCR; s <<= 1) {
        float na[PER_T], nd[PER_T], ne[PER_T], nr[PER_T];
#pragma unroll
        for (int k = 0; k < PER_T; ++k) {
            int i = t + k * TPB;
            float ai = sa[i], di = sd[i], ei = se[i], ri = sr[i];
            float nav = 0.0f, nev = 0.0f, ndv = di, nrv = ri;
            int im = i - s;
            if (im >= 0) {
                float k1 = ai / sd[im];
                nav  = -k1 * sa[im];
                ndv -=  k1 * se[im];
                nrv -=  k1 * sr[im];
            }
            int ip = i + s;
            if (ip < NPCR) {
                float k2 = ei / sd[ip];
                nev  = -k2 * se[ip];
                ndv -=  k2 * sa[ip];
                nrv -=  k2 * sr[ip];
            }
            na[k] = nav; nd[k] = ndv; ne[k] = nev; nr[k] = nrv;
        }
        __syncthreads();
#pragma unroll
        for (int k = 0; k < PER_T; ++k) {
            int i = t + k * TPB;
            sa[i] = na[k]; sd[i] = nd[k]; se[i] = ne[k]; sr[i] = nr[k];
        }
        __syncthreads();
    }

    // ---- Fully decoupled: y_i = r_i / d_i (in place into sr)
#pragma unroll
    for (int k = 0; k < PER_T; ++k) {
        int i = t + k * TPB;
        sr[i] = sr[i] / sd[i];
    }
    __syncthreads();

    // ---- x = A^T y, scaled: x[0]=c*y[0]; x[i]=y[i-1]+c*y[i]; x[N]=y[N-1]
    for (int i = t; i <= NPCR; i += TPB) {
        float x;
        if (i == 0)          x = c * sr[0];
        else if (i == NPCR)  x = sr[NPCR - 1];
        else                 x = sr[i - 1] + c * sr[i];
        out[i] = x * istd;
    }
}

extern "C" void kernel_launch(void* const* d_in, const int* in_sizes, int n_in,
                              void* d_out, int out_size, void* d_ws, size_t ws_size,
                              hipStream_t stream) {
    (void)in_sizes; (void)n_in; (void)out_size; (void)d_ws; (void)ws_size;
    const float* time_block = (const float*)d_in[0];
    const float* arma_const = (const float*)d_in[1];
    const float* ma_coeff   = (const float*)d_in[2];
    const float* std_innov  = (const float*)d_in[3];
    float* out = (float*)d_out;

    arima_ma1_innov_kernel<<<1, TPB, 0, stream>>>(
        time_block, arma_const, ma_coeff, std_innov, out);
}